// OKT_74268574483046
// MI455X (gfx1250) — compile-verified
//
#include <hip/hip_runtime.h>
#include <hip/hip_bf16.h>
#include <math.h>

typedef float v2f __attribute__((ext_vector_type(2)));
typedef float v8f __attribute__((ext_vector_type(8)));

#define KROWS 50000
#define EK    128
#define HH    512
#define NOUT  1536          // 3*H
#define DIN   513
#define MB    64            // rows of h2 per workgroup in main GEMM
#define LDA   516           // padded LDS row stride (floats): bank-conflict-free frag loads
#define NB_MAIN ((KROWS + MB - 1) / MB)        // 782
#define NB_K   ((KROWS + 255) / 256)           // 196

// ---- workspace layout (floats) ----
#define OFF_ALPHA   0
#define OFF_E       (OFF_ALPHA + KROWS)
#define OFF_BMAX    (OFF_E + KROWS)
#define OFF_BSUM    (OFF_BMAX + 256)
#define OFF_SCAL    (OFF_BSUM + 256)          // [0]=max, [1]=1/sumExp
#define OFF_GX      (OFF_SCAL + 4)
#define OFF_HKP     (OFF_GX + NOUT)
#define OFF_HKPPART (OFF_HKP + HH)            // NB_MAIN * 512

__device__ __forceinline__ v8f wmma_f32(v2f a, v2f b, v8f c) {
    return __builtin_amdgcn_wmma_f32_16x16x4_f32(false, a, false, b, (short)0, c,
                                                 false, false);
}
// sigmoid via single v_rcp_f32: exp overflow -> inf -> rcp -> 0 gives the correct limit
__device__ __forceinline__ float sigmoidf_(float x) {
    return __builtin_amdgcn_rcpf(1.0f + __expf(-x));
}
// tanh via exp + rcp; clamp avoids inf/inf NaN (tanh saturates in fp32 by |x|~9)
__device__ __forceinline__ float tanhf_(float x) {
    x = fminf(fmaxf(x, -20.0f), 20.0f);
    float t = __expf(-2.0f * x);
    return (1.0f - t) * __builtin_amdgcn_rcpf(1.0f + t);
}

// ---- K1: alpha[k] = KM[k,:] . o_e ; per-block max ----
__global__ void k_alpha(const float* __restrict__ km, const float* __restrict__ oe,
                        float* __restrict__ alpha, float* __restrict__ bmax) {
    __shared__ float soe[EK];
    __shared__ float red[256];
    int tid = threadIdx.x;
    if (tid < EK) soe[tid] = oe[tid];
    __syncthreads();
    int k = blockIdx.x * 256 + tid;
    float a = -3.4e38f;
    if (k < KROWS) {
        const float4* row = (const float4*)(km + (size_t)k * EK);
        float acc = 0.f;
#pragma unroll
        for (int i = 0; i < EK / 4; ++i) {
            float4 v = row[i];
            const float* o = soe + i * 4;
            acc += v.x * o[0] + v.y * o[1] + v.z * o[2] + v.w * o[3];
        }
        alpha[k] = acc;
        a = acc;
    }
    red[tid] = a;
    __syncthreads();
    for (int s = 128; s > 0; s >>= 1) {
        if (tid < s) red[tid] = fmaxf(red[tid], red[tid + s]);
        __syncthreads();
    }
    if (tid == 0) bmax[blockIdx.x] = red[0];
}

// ---- K2: global max ----
__global__ void k_maxred(const float* __restrict__ bmax, int n, float* __restrict__ scal) {
    __shared__ float red[256];
    int tid = threadIdx.x;
    float a = -3.4e38f;
    for (int i = tid; i < n; i += 256) a = fmaxf(a, bmax[i]);
    red[tid] = a;
    __syncthreads();
    for (int s = 128; s > 0; s >>= 1) {
        if (tid < s) red[tid] = fmaxf(red[tid], red[tid + s]);
        __syncthreads();
    }
    if (tid == 0) scal[0] = red[0];
}

// ---- K3: e[k] = exp(alpha-max); per-block sum ----
__global__ void k_exp(const float* __restrict__ alpha, const float* __restrict__ scal,
                      float* __restrict__ e, float* __restrict__ bsum) {
    __shared__ float red[256];
    int tid = threadIdx.x;
    float mx = scal[0];
    int k = blockIdx.x * 256 + tid;
    float v = 0.f;
    if (k < KROWS) {
        v = __expf(alpha[k] - mx);
        e[k] = v;
    }
    red[tid] = v;
    __syncthreads();
    for (int s = 128; s > 0; s >>= 1) {
        if (tid < s) red[tid] += red[tid + s];
        __syncthreads();
    }
    if (tid == 0) bsum[blockIdx.x] = red[0];
}

// ---- K4: 1/sum ----
__global__ void k_sumred(const float* __restrict__ bsum, int n, float* __restrict__ scal) {
    __shared__ float red[256];
    int tid = threadIdx.x;
    float a = 0.f;
    for (int i = tid; i < n; i += 256) a += bsum[i];
    red[tid] = a;
    __syncthreads();
    for (int s = 128; s > 0; s >>= 1) {
        if (tid < s) red[tid] += red[tid + s];
        __syncthreads();
    }
    if (tid == 0) scal[1] = 1.0f / red[0];
}

// ---- K5: gx = W_ih @ [ex_e, s]  (1536 values) ----
__global__ void k_gx(const float* __restrict__ wih, const float* __restrict__ exe,
                     const float* __restrict__ s, float* __restrict__ gx) {
    int j = blockIdx.x * 256 + threadIdx.x;
    if (j >= NOUT) return;
    const float* row = wih + (size_t)j * DIN;
    float acc = row[HH] * s[0];
    for (int c = 0; c < HH; ++c) acc += row[c] * exe[c];
    gx[j] = acc;
}

// gate epilogue for one triple; GUARD=false for full blocks (781 of 782)
template <bool GUARD>
__device__ __forceinline__ void gru_epilogue(
    const v8f* accR, const v8f* accZ, const v8f* accN,
    const float* lA, const float* lBeta, float* __restrict__ out,
    int rowBase, int j, int lhalf,
    float gxr, float bir, float bhr,
    float gxz, float biz, float bhz,
    float gxn, float bin_, float bhn) {
#pragma unroll
    for (int m = 0; m < 4; ++m) {
#pragma unroll
        for (int v = 0; v < 8; ++v) {
            int lrow = m * 16 + v + 8 * lhalf;   // C/D layout: lanes>=16 hold M=v+8
            int k = rowBase + lrow;
            float bk = lBeta[lrow];
            float r = sigmoidf_(bk * gxr + bir + bhr + accR[m][v]);
            float z = sigmoidf_(bk * gxz + biz + bhz + accZ[m][v]);
            float nn = tanhf_(bk * gxn + bin_ + r * (accN[m][v] + bhn));
            float hp = lA[lrow * LDA + j];
            float hnew = (1.f - z) * nn + z * hp;
            if (GUARD) {
                if (k < KROWS) out[1 + (size_t)k * HH + j] = hnew;
            } else {
                out[1 + (size_t)k * HH + j] = hnew;
            }
        }
    }
}

// ---- K6: main fused kernel: gh = h2 @ W_hh^T (WMMA fp32) + GRU gates + hkp partials ----
__global__ void __launch_bounds__(256)
k_main(const float* __restrict__ h2, const float* __restrict__ whh,
       const float* __restrict__ bih, const float* __restrict__ bhh,
       const float* __restrict__ e, const float* __restrict__ scal,
       const float* __restrict__ gx, float* __restrict__ out,
       float* __restrict__ hkpPart) {
    extern __shared__ float smem[];
    float* lA = smem;                 // [MB][LDA]
    float* lBeta = smem + MB * LDA;   // [MB]

    const int tid = threadIdx.x;
    const int rowBase = blockIdx.x * MB;
    const bool fullTile = (rowBase + MB <= KROWS);   // uniform across block
    const float invS = scal[1];

    // stage h2 tile (zero-fill out-of-range rows)
    for (int i = tid; i < MB * (HH / 4); i += 256) {
        int r = i >> 7;         // HH/4 = 128 float4 per row
        int c4 = i & 127;
        int grow = rowBase + r;
        float4 v = make_float4(0.f, 0.f, 0.f, 0.f);
        if (grow < KROWS) v = ((const float4*)(h2 + (size_t)grow * HH))[c4];
        float* dst = lA + r * LDA + c4 * 4;
        dst[0] = v.x; dst[1] = v.y; dst[2] = v.z; dst[3] = v.w;
    }
    if (tid < MB) {
        int grow = rowBase + tid;
        lBeta[tid] = (grow < KROWS) ? e[grow] * invS : 0.f;
    }
    __syncthreads();

    // hkp partial: sum_r beta_r * h2[r, j]  (deterministic per-block partials)
    for (int j = tid; j < HH; j += 256) {
        float acc = 0.f;
#pragma unroll 8
        for (int r = 0; r < MB; ++r) acc += lBeta[r] * lA[r * LDA + j];
        hkpPart[(size_t)blockIdx.x * HH + j] = acc;
    }

    const int wave = tid >> 5;
    const int lane = tid & 31;
    const int lhalf = lane >> 4;   // 0 or 1
    const int l16 = lane & 15;
    const int coff = 2 * lhalf;    // K sub-offset within a frag

    // 32 column-triples (j, j+512, j+1024) of 16 cols each, split across 8 waves
    for (int tri = wave; tri < 32; tri += 8) {
        v8f accR[4], accZ[4], accN[4];
#pragma unroll
        for (int m = 0; m < 4; ++m) { accR[m] = (v8f){}; accZ[m] = (v8f){}; accN[m] = (v8f){}; }

        const int nrow = tri * 16 + l16;                    // W_hh row for this lane
        const float* bR = whh + (size_t)nrow * HH;
        const float* bZ = bR + (size_t)HH * HH;             // +512 rows
        const float* bN = bR + (size_t)2 * HH * HH;         // +1024 rows

        // hoist epilogue scalars: latency hides behind the GEMM loop
        const int j = tri * 16 + l16;
        const float gxr = gx[j],        bir = bih[j],        bhr = bhh[j];
        const float gxz = gx[j + HH],   biz = bih[j + HH],   bhz = bhh[j + HH];
        const float gxn = gx[j + 2*HH], bin_ = bih[j + 2*HH], bhn = bhh[j + 2*HH];

        for (int kk = 0; kk < HH; kk += 4) {
            v2f bfR = *(const v2f*)(bR + kk + coff);
            v2f bfZ = *(const v2f*)(bZ + kk + coff);
            v2f bfN = *(const v2f*)(bN + kk + coff);
#pragma unroll
            for (int m = 0; m < 4; ++m) {
                v2f af = *(const v2f*)(lA + (m * 16 + l16) * LDA + kk + coff);
                accR[m] = wmma_f32(af, bfR, accR[m]);
                accZ[m] = wmma_f32(af, bfZ, accZ[m]);
                accN[m] = wmma_f32(af, bfN, accN[m]);
            }
        }

        if (fullTile) {
            gru_epilogue<false>(accR, accZ, accN, lA, lBeta, out, rowBase, j, lhalf,
                                gxr, bir, bhr, gxz, biz, bhz, gxn, bin_, bhn);
        } else {
            gru_epilogue<true>(accR, accZ, accN, lA, lBeta, out, rowBase, j, lhalf,
                               gxr, bir, bhr, gxz, biz, bhz, gxn, bin_, bhn);
        }
    }
}

// ---- K7: hkp[j] = invS * sum_blocks part[b][j] ----
__global__ void k_hkp(const float* __restrict__ part, const float* __restrict__ scal,
                      float* __restrict__ hkp, int nblk) {
    __shared__ float red[128];
    int j = blockIdx.x;
    int tid = threadIdx.x;
    float a = 0.f;
    for (int b = tid; b < nblk; b += 128) a += part[(size_t)b * HH + j];
    red[tid] = a;
    __syncthreads();
    for (int s = 64; s > 0; s >>= 1) {
        if (tid < s) red[tid] += red[tid + s];
        __syncthreads();
    }
    if (tid == 0) hkp[j] = red[0] * scal[1];
}

// ---- K8: predict_score = score_W . [ex_e, hkp] + score_b ----
__global__ void k_score(const float* __restrict__ sw, const float* __restrict__ sb,
                        const float* __restrict__ exe, const float* __restrict__ hkp,
                        float* __restrict__ out) {
    __shared__ float red[256];
    int tid = threadIdx.x;
    float a = 0.f;
    for (int jj = tid; jj < HH + HH; jj += 256) {
        float v = (jj < HH) ? exe[jj] : hkp[jj - HH];
        a += sw[jj] * v;
    }
    red[tid] = a;
    __syncthreads();
    for (int s = 128; s > 0; s >>= 1) {
        if (tid < s) red[tid] += red[tid + s];
        __syncthreads();
    }
    if (tid == 0) out[0] = red[0] + sb[0];
}

extern "C" void kernel_launch(void* const* d_in, const int* in_sizes, int n_in,
                              void* d_out, int out_size, void* d_ws, size_t ws_size,
                              hipStream_t stream) {
    (void)in_sizes; (void)n_in; (void)out_size; (void)ws_size;
    const float* o_e  = (const float*)d_in[0];
    const float* ex_e = (const float*)d_in[1];
    const float* s    = (const float*)d_in[2];
    const float* h    = (const float*)d_in[3];
    const float* km   = (const float*)d_in[4];
    const float* W_ih = (const float*)d_in[5];
    const float* W_hh = (const float*)d_in[6];
    const float* b_ih = (const float*)d_in[7];
    const float* b_hh = (const float*)d_in[8];
    const float* sw   = (const float*)d_in[9];
    const float* sb   = (const float*)d_in[10];
    float* out = (float*)d_out;
    float* ws  = (float*)d_ws;

    float* alpha = ws + OFF_ALPHA;
    float* e     = ws + OFF_E;
    float* bmax  = ws + OFF_BMAX;
    float* bsum  = ws + OFF_BSUM;
    float* scal  = ws + OFF_SCAL;
    float* gx    = ws + OFF_GX;
    float* hkp   = ws + OFF_HKP;
    float* hpart = ws + OFF_HKPPART;

    k_alpha<<<NB_K, 256, 0, stream>>>(km, o_e, alpha, bmax);
    k_maxred<<<1, 256, 0, stream>>>(bmax, NB_K, scal);
    k_exp<<<NB_K, 256, 0, stream>>>(alpha, scal, e, bsum);
    k_sumred<<<1, 256, 0, stream>>>(bsum, NB_K, scal);
    k_gx<<<(NOUT + 255) / 256, 256, 0, stream>>>(W_ih, ex_e, s, gx);

    size_t lds_bytes = (size_t)(MB * LDA + MB) * sizeof(float);   // ~132 KB
    hipFuncSetAttribute((const void*)k_main,
                        hipFuncAttributeMaxDynamicSharedMemorySize,
                        (int)lds_bytes);
    k_main<<<NB_MAIN, 256, lds_bytes, stream>>>(h, W_hh, b_ih, b_hh, e, scal, gx,
                                                out, hpart);

    k_hkp<<<HH, 128, 0, stream>>>(hpart, scal, hkp, NB_MAIN);
    k_score<<<1, 256, 0, stream>>>(sw, sb, ex_e, hkp, out);
}